// two_d_CXN_CMPS_19696720019795
// MI455X (gfx1250) — compile-verified
//
#include <hip/hip_runtime.h>

typedef __attribute__((ext_vector_type(2))) float v2f;
typedef __attribute__((ext_vector_type(4))) float v4f;
typedef __attribute__((ext_vector_type(8))) float v8f;

// ---------------------------------------------------------------------------
// Kernel 1: H = X @ W   (X:[n,32], W:[32,32]) -> H written in the WMMA
// B-fragment interleaved order:  H_frag[(r>>1)*64 + c*2 + (r&1)] = H[r][c]
// so that a K-chunk of B rows is one contiguous block of floats.
// ---------------------------------------------------------------------------
__global__ void xw_kernel(const float* __restrict__ X, const float* __restrict__ W,
                          float* __restrict__ H, int nrows) {
  int t = blockIdx.x * blockDim.x + threadIdx.x;
  if (t >= nrows * 32) return;
  int r = t >> 5, c = t & 31;
  const float* xr = X + (size_t)r * 32;
  float acc = 0.f;
#pragma unroll
  for (int k = 0; k < 32; ++k) acc = fmaf(xr[k], W[k * 32 + c], acc);
  H[((size_t)(r >> 1) * 64) + c * 2 + (r & 1)] = acc;
}

// ---------------------------------------------------------------------------
// Kernel 2: Z[M,32] = G0[M,K0] @ H0  (+ G1[M,K1] @ H1), H pre-swizzled.
// 256 threads = 8 waves; each wave computes a 16x32 output tile with
// v_wmma_f32_16x16x4_f32 (two N-halves). G streamed with non-temporal b64
// loads in native A-fragment layout; B chunk staged through LDS per block.
// ---------------------------------------------------------------------------
__launch_bounds__(256)
__global__ void gh_kernel(const float* __restrict__ G0, const float* __restrict__ H0, int K0,
                          const float* __restrict__ G1, const float* __restrict__ H1, int K1,
                          float* __restrict__ Z, int M) {
  __shared__ float lds_b[1024];           // 32 K-rows x 32 N, interleaved pairs (4 KB)

  const int tid   = threadIdx.x;
  const int lane  = tid & 31;
  const int wave  = tid >> 5;
  const int nrow  = lane & 15;            // A-frag M row index == B-frag N col index
  const int khalf = lane >> 4;            // 0 -> K pair +0, 1 -> K pair +2

  int m0 = blockIdx.x * 128 + wave * 16;
  if (m0 > M - 16) m0 = M - 16;           // M is a multiple of 16; clamp tail waves

  v8f c0 = {};                            // N columns 0..15
  v8f c1 = {};                            // N columns 16..31

  for (int src = 0; src < 2; ++src) {
    const float* G = src ? G1 : G0;
    const float* H = src ? H1 : H0;
    const int    K = src ? K1 : K0;
    if (K <= 0) continue;
    const float* grow = G + (size_t)(m0 + nrow) * (size_t)K;   // this lane's A row

    for (int k0 = 0; k0 < K; k0 += 32) {
      int kc = K - k0; if (kc > 32) kc = 32;                   // 32 or 16 (K%32)
      __syncthreads();                                         // WAR on lds_b
      if (tid * 4 < kc * 32) {                                 // stage B chunk
        const v4f* s4 = (const v4f*)(H + (size_t)k0 * 32);
        ((v4f*)lds_b)[tid] = s4[tid];
      }
      __syncthreads();

      auto wstep = [&](int s) {
        const int kb = s * 4 + khalf * 2;                      // even -> 8B aligned
        // A fragment: lanes 0-15 hold (row, K+{0,1}); lanes 16-31 (row, K+{2,3})
        v2f a = __builtin_nontemporal_load((const v2f*)(grow + k0 + kb));
        const float* bp = lds_b + ((s * 2 + khalf) << 6);      // pair-row base
        v2f b0 = *(const v2f*)(bp + 2 * nrow);
        v2f b1 = *(const v2f*)(bp + 2 * (nrow + 16));
        c0 = __builtin_amdgcn_wmma_f32_16x16x4_f32(false, a, false, b0,
                                                   (short)0, c0, false, false);
        c1 = __builtin_amdgcn_wmma_f32_16x16x4_f32(false, a, false, b1,
                                                   (short)0, c1, false, false);
      };
      if (kc == 32) {
#pragma unroll
        for (int s = 0; s < 8; ++s) wstep(s);
      } else {
#pragma unroll
        for (int s = 0; s < 4; ++s) wstep(s);
      }
    }
  }

  // D layout: VGPR j, lanes 0-15 -> (M=j,   N=lane),  lanes 16-31 -> (M=j+8, N=lane-16)
  float* zb = Z + (size_t)(m0 + khalf * 8) * 32 + nrow;
#pragma unroll
  for (int j = 0; j < 8; ++j) {
    zb[(size_t)j * 32]      = c0[j];
    zb[(size_t)j * 32 + 16] = c1[j];
  }
}

// ---------------------------------------------------------------------------
extern "C" void kernel_launch(void* const* d_in, const int* in_sizes, int n_in,
                              void* d_out, int out_size, void* d_ws, size_t ws_size,
                              hipStream_t stream) {
  const float* xv   = (const float*)d_in[0];
  const float* xe   = (const float*)d_in[1];
  const float* xf   = (const float*)d_in[2];
  const float* Gv2v = (const float*)d_in[3];
  const float* Gv2e = (const float*)d_in[4];
  const float* Ge2e = (const float*)d_in[5];
  const float* Ge2f = (const float*)d_in[6];
  const float* Gf2f = (const float*)d_in[7];
  const float* Wv2v = (const float*)d_in[8];
  const float* Wv2e = (const float*)d_in[9];
  const float* We2e = (const float*)d_in[10];
  const float* We2f = (const float*)d_in[11];
  const float* Wf2f = (const float*)d_in[12];

  float* out = (float*)d_out;
  float* ws  = (float*)d_ws;

  const int NV = 6000, NE = 12000, NF = 8000;

  // workspace layout (floats): 5 H matrices, fragment-ordered. 5.63 MB total.
  float* hv2v = ws;
  float* hv2e = hv2v + (size_t)NV * 32;
  float* he2e = hv2e + (size_t)NV * 32;
  float* he2f = he2e + (size_t)NE * 32;
  float* hf2f = he2f + (size_t)NE * 32;

  auto xw = [&](const float* X, const float* W, float* H, int n) {
    int t = n * 32;
    xw_kernel<<<(t + 255) / 256, 256, 0, stream>>>(X, W, H, n);
  };
  xw(xv, Wv2v, hv2v, NV);
  xw(xv, Wv2e, hv2e, NV);
  xw(xe, We2e, he2e, NE);
  xw(xe, We2f, he2f, NE);
  xw(xf, Wf2f, hf2f, NF);

  float* zv = out;
  float* ze = zv + (size_t)NV * 32;
  float* zf = ze + (size_t)NE * 32;

  // zv = Gv2v @ hv2v
  gh_kernel<<<(NV + 127) / 128, 256, 0, stream>>>(Gv2v, hv2v, NV,
                                                  nullptr, nullptr, 0, zv, NV);
  // ze = Gv2e @ hv2e + Ge2e @ he2e      (Gv2e: [NE,NV], Ge2e: [NE,NE])
  gh_kernel<<<(NE + 127) / 128, 256, 0, stream>>>(Gv2e, hv2e, NV,
                                                  Ge2e, he2e, NE, ze, NE);
  // zf = Ge2f @ he2f + Gf2f @ hf2f      (Ge2f: [NF,NE], Gf2f: [NF,NF])
  gh_kernel<<<(NF + 127) / 128, 256, 0, stream>>>(Ge2f, he2f, NE,
                                                  Gf2f, hf2f, NF, zf, NF);
}